// GraphConv_79242146611301
// MI455X (gfx1250) — compile-verified
//
#include <hip/hip_runtime.h>

// GraphConv SpMM: out[r,:] = sum_{e: rows[e]==r} vals[e] * emb[cols[e],:]
// N=100000 nodes, E=1600000 edges, D=64.
//
// Roofline: 2*E*D = 0.2 GFLOP vs ~70MB of HBM traffic (19.2MB edge arrays +
// 25.6MB embedding table + 25.6MB output) -> ~3us floor at 23.3 TB/s. Purely
// memory/atomic bound; WMMA has no dense tile structure to exploit in
// unsorted COO. MI455X-specific wins (verified in disasm via probe):
//   - unsafeAtomicAdd -> native global_atomic_add_f32 (no-return, SCOPE_DEV)
//     serviced by L2 atomic units: the 25.6MB output is L2-resident (192MB L2).
//   - 16 lanes x float4 per edge -> coalesced global_load_b128 gathers of
//     256B embedding rows (table also L2-resident).
//   - __builtin_prefetch -> gfx1250 global_prefetch_b8, issued one grid-stride
//     iteration ahead to hide the data-dependent gather latency.

#define DIM 64
#define VEC 4                         // float4 per lane
#define LANES_PER_EDGE (DIM / VEC)    // 16 lanes cover one 64-float row
#define BLOCK 256

__global__ __launch_bounds__(BLOCK)
void spmm_scatter_kernel(const float* __restrict__ emb,
                         const float* __restrict__ vals,
                         const int*   __restrict__ rows,
                         const int*   __restrict__ cols,
                         float*       __restrict__ out,
                         int nEdges) {
    const int gtid   = blockIdx.x * BLOCK + threadIdx.x;
    const int slot   = gtid >> 4;                       // edge slot (16 lanes/edge)
    const int seg    = gtid & (LANES_PER_EDGE - 1);     // which float4 chunk of row
    const int nSlots = (gridDim.x * BLOCK) >> 4;

    if (slot >= nEdges) return;

    // Column index for the current edge; carried across iterations so each
    // iteration issues exactly one cols[] load (for the prefetch target).
    int col = cols[slot];

    for (int e = slot; e < nEdges; e += nSlots) {
        const int en = e + nSlots;
        int colNext = 0;
        if (en < nEdges) {
            colNext = cols[en];
            // gfx1250 global_prefetch_b8: warm next iteration's gathered row.
            __builtin_prefetch(emb + (size_t)colNext * DIM + seg * VEC, 0, 1);
        }

        const int   row = rows[e];
        const float v   = vals[e];

        // 16 lanes x float4 = one 64-float embedding row, global_load_b128.
        const float4 m =
            *reinterpret_cast<const float4*>(emb + (size_t)col * DIM + seg * VEC);

        float* dst = out + (size_t)row * DIM + seg * VEC;
        // Native global_atomic_add_f32 (no-return) into L2-resident output.
        unsafeAtomicAdd(dst + 0, v * m.x);
        unsafeAtomicAdd(dst + 1, v * m.y);
        unsafeAtomicAdd(dst + 2, v * m.z);
        unsafeAtomicAdd(dst + 3, v * m.w);

        col = colNext;
    }
}

__global__ __launch_bounds__(BLOCK)
void zero_out_kernel(float4* __restrict__ out, int n4) {
    int i = blockIdx.x * BLOCK + threadIdx.x;
    if (i < n4) out[i] = make_float4(0.f, 0.f, 0.f, 0.f);
}

extern "C" void kernel_launch(void* const* d_in, const int* in_sizes, int n_in,
                              void* d_out, int out_size, void* d_ws, size_t ws_size,
                              hipStream_t stream) {
    const float* emb  = (const float*)d_in[0];   // [N, 64] fp32
    const float* vals = (const float*)d_in[1];   // [E] fp32
    const int*   rows = (const int*)d_in[2];     // [E] i32
    const int*   cols = (const int*)d_in[3];     // [E] i32
    float*       out  = (float*)d_out;           // [N, 64] fp32

    const int nEdges = in_sizes[1];

    // 1) Zero the (poisoned) output before atomic accumulation.
    const int n4 = out_size / 4;
    zero_out_kernel<<<(n4 + BLOCK - 1) / BLOCK, BLOCK, 0, stream>>>(
        (float4*)out, n4);

    // 2) Edge-parallel scatter. ~4 grid-stride iterations per thread so the
    //    one-iteration-ahead prefetch has useful distance.
    const long long laneTasks = (long long)nEdges * LANES_PER_EDGE;
    const int iters  = 4;
    int blocks = (int)((laneTasks + (long long)BLOCK * iters - 1) /
                       ((long long)BLOCK * iters));
    if (blocks < 1) blocks = 1;
    spmm_scatter_kernel<<<blocks, BLOCK, 0, stream>>>(
        emb, vals, rows, cols, out, nEdges);
}